// HEM_1803886265834
// MI455X (gfx1250) — compile-verified
//
#include <hip/hip_runtime.h>
#include <cstdint>

// Problem constants (16, 3, 720, 1280) f32
#define BB       16
#define CCH      3
#define HGT      720
#define WID      1280
#define HW       (HGT * WID)          // 921600
#define HARD_IND 460800u              // int(0.5 * hw)
#define NTOT     (BB * CCH * HW)      // 44236800
#define RAND_CUT 429496730u           // ~0.1 * 2^32

typedef float v2f __attribute__((ext_vector_type(2)));
typedef float v4f __attribute__((ext_vector_type(4)));
typedef float v8f __attribute__((ext_vector_type(8)));

// ---------------------------------------------------------------------------
// deterministic per-(batch, pixel) pseudo-random 10% selection
// ---------------------------------------------------------------------------
__device__ __forceinline__ bool rand_sel(unsigned b, unsigned i) {
  unsigned h = (b + 1u) * 0x9E3779B9u ^ (i * 0x85EBCA6Bu);
  h ^= h >> 16; h *= 0x7FEB352Du;
  h ^= h >> 15; h *= 0x846CA68Bu;
  h ^= h >> 16;
  return h < RAND_CUT;
}

// Low 32 bits of a generic pointer into LDS == workgroup-relative LDS byte
// address (ISA 10.2: LDS aperture maps addr[31:0] -> LDS offset).
__device__ __forceinline__ unsigned lds_off_of(const void* p) {
  return (unsigned)(uintptr_t)p;
}

// ---------------------------------------------------------------------------
// K1: res = sum_c |x-y|  (nontemporal 128-bit streaming loads: x,y are 353MB,
// bigger than the 192MB L2 -> bypass; res (59MB) stays temporal so later
// passes hit L2).  Also builds coarse per-batch histogram of res bits[31:21].
// ---------------------------------------------------------------------------
__global__ __launch_bounds__(256) void k_res_hist(const float* __restrict__ x,
                                                  const float* __restrict__ y,
                                                  float* __restrict__ res,
                                                  unsigned* __restrict__ hist1) {
  __shared__ unsigned lh[2048];
  const int tid = threadIdx.x;
  const int b   = blockIdx.y;
  for (int i = tid; i < 2048; i += 256) lh[i] = 0u;
  __syncthreads();

  const int    pix = (blockIdx.x * 256 + tid) * 4;   // 4 pixels / thread
  const size_t xb  = (size_t)b * (CCH * HW);

  v4f r = {0.f, 0.f, 0.f, 0.f};
#pragma unroll
  for (int c = 0; c < CCH; ++c) {
    v4f a  = __builtin_nontemporal_load((v4f*)(x + xb + (size_t)c * HW + pix));
    v4f bv = __builtin_nontemporal_load((v4f*)(y + xb + (size_t)c * HW + pix));
    v4f d = a - bv;
    r.x += fabsf(d.x); r.y += fabsf(d.y); r.z += fabsf(d.z); r.w += fabsf(d.w);
  }
  *(v4f*)(res + (size_t)b * HW + pix) = r;

  // res >= 0  =>  float-bit order == uint order; bucket on top 11 bits
  atomicAdd(&lh[__float_as_uint(r.x) >> 21], 1u);
  atomicAdd(&lh[__float_as_uint(r.y) >> 21], 1u);
  atomicAdd(&lh[__float_as_uint(r.z) >> 21], 1u);
  atomicAdd(&lh[__float_as_uint(r.w) >> 21], 1u);
  __syncthreads();

  unsigned* gh = hist1 + b * 2048;
  for (int i = tid; i < 2048; i += 256) {
    unsigned v = lh[i];
    if (v) atomicAdd(&gh[i], v);
  }
}

// ---------------------------------------------------------------------------
// Radix-select scan: walk histogram from the top bin, find bin containing
// descending rank k; update (prefix, remaining rank).  Tiny, 1 block/batch.
// ---------------------------------------------------------------------------
__global__ void k_scan(const unsigned* __restrict__ hist,
                       unsigned* __restrict__ state,
                       int nbins, int shift, int stage) {
  if (threadIdx.x != 0) return;
  const int b = blockIdx.x;
  unsigned k      = (stage == 0) ? HARD_IND : state[b * 2 + 1];
  unsigned prefix = (stage == 0) ? 0u       : state[b * 2 + 0];
  const unsigned* h = hist + b * 2048;
  unsigned cum = 0; int sel = 0;
  for (int bin = nbins - 1; bin >= 0; --bin) {
    unsigned c = h[bin];
    if (cum + c > k) { sel = bin; break; }
    cum += c;
  }
  state[b * 2 + 0] = prefix | ((unsigned)sel << shift);
  state[b * 2 + 1] = k - cum;
}

// ---------------------------------------------------------------------------
// K2/K3: refinement histograms over res (L2-resident), filtered by prefix.
// stage==1: filter bits[31:21], bucket bits[20:10]; stage==2: filter
// bits[31:10], bucket bits[9:0].
// ---------------------------------------------------------------------------
__global__ __launch_bounds__(256) void k_hist_refine(const float* __restrict__ res,
                                                     const unsigned* __restrict__ state,
                                                     unsigned* __restrict__ histo,
                                                     int stage) {
  __shared__ unsigned lh[2048];
  const int tid = threadIdx.x;
  const int b   = blockIdx.y;
  const int nb  = (stage == 1) ? 2048 : 1024;
  for (int i = tid; i < 2048; i += 256) lh[i] = 0u;
  __syncthreads();

  const unsigned prefix = state[b * 2 + 0];
  const int pix = (blockIdx.x * 256 + tid) * 4;
  const v4f r = *(const v4f*)(res + (size_t)b * HW + pix);
#pragma unroll
  for (int e = 0; e < 4; ++e) {
    const unsigned bits = __float_as_uint(r[e]);
    if (stage == 1) {
      if ((bits >> 21) == (prefix >> 21)) atomicAdd(&lh[(bits >> 10) & 2047u], 1u);
    } else {
      if ((bits >> 10) == (prefix >> 10)) atomicAdd(&lh[bits & 1023u], 1u);
    }
  }
  __syncthreads();

  unsigned* gh = histo + b * 2048;
  for (int i = tid; i < nb; i += 256) {
    unsigned v = lh[i];
    if (v) atomicAdd(&gh[i], v);
  }
}

// ---------------------------------------------------------------------------
// K4: masked sum.  res tiles are staged global->LDS with the CDNA5 async data
// mover (global_load_async_to_lds_b128, ASYNCcnt) in a double buffer; each
// wave copies & consumes only its own 512-float slice, so no barriers needed.
// Masked values are accumulated on the matrix pipe: C += A x ones via
// V_WMMA_F32_16X16X4_F32 (B = all-ones makes the result layout-independent:
// every input appears exactly once in A, so the grand total is the column
// sum; each D column is identical).  Keeps the VALU free for address math /
// mask hashing and removes the serial FP add chain.  EXEC is all-ones here
// (masking is a select, control flow is wave-uniform) as WMMA requires.
// ---------------------------------------------------------------------------
#define TILE          4096     // floats per block-tile (16 KB)
#define TILES_PER_BLK 5        // 45 blocks/batch * 5 * 4096 = 921600

__global__ __launch_bounds__(256) void k_masked_sum(const float* __restrict__ res,
                                                    const unsigned* __restrict__ state,
                                                    float* __restrict__ acc) {
  __shared__ __align__(16) float lbuf[2][TILE];   // 32 KB double buffer
  const int tid  = threadIdx.x;
  const int lane = tid & 31;
  const int w    = tid >> 5;                      // wave id, 0..7 (wave32)
  const int b    = blockIdx.y;
  const unsigned thre = state[b * 2 + 0];         // exact threshold bits
  const float* rbase  = res + (size_t)b * HW;
  const int tile0     = blockIdx.x * TILES_PER_BLK;

  auto issue = [&](int t, int buf) {
    const int base = (tile0 + t) * TILE + w * 512 + lane * 4;
#pragma unroll
    for (int it = 0; it < 4; ++it) {
      const float* g = rbase + base + it * 128;
      unsigned loff  = lds_off_of(&lbuf[buf][w * 512 + lane * 4 + it * 128]);
      asm volatile("global_load_async_to_lds_b128 %0, %1, off"
                   :: "v"(loff), "v"(g) : "memory");
    }
  };

  v8f cacc = {0.f, 0.f, 0.f, 0.f, 0.f, 0.f, 0.f, 0.f};
  const v2f bones = {1.f, 1.f};                   // B = ones(4x16)

  issue(0, 0);
  for (int t = 0; t < TILES_PER_BLK; ++t) {
    if (t + 1 < TILES_PER_BLK) {
      issue(t + 1, (t + 1) & 1);
      // 4 async ops in flight for next tile; wait until previous tile's 4 done
      asm volatile("s_wait_asynccnt 4" ::: "memory");
    } else {
      asm volatile("s_wait_asynccnt 0" ::: "memory");
    }
    const int    ibase = (tile0 + t) * TILE + w * 512 + lane * 4;
    const float* lb    = &lbuf[t & 1][w * 512 + lane * 4];
#pragma unroll
    for (int it = 0; it < 4; ++it) {
      v4f r = *(const v4f*)(lb + it * 128);
      v4f m;
#pragma unroll
      for (int e = 0; e < 4; ++e) {
        unsigned idx  = (unsigned)(ibase + it * 128 + e);
        unsigned bits = __float_as_uint(r[e]);
        bool hard = bits > thre;                 // strict >, matches reference
        bool rnd  = rand_sel((unsigned)b, idx);
        m[e] = (hard | rnd) ? r[e] : 0.f;        // select keeps EXEC all-ones
      }
      v2f a0 = {m.x, m.y};
      v2f a1 = {m.z, m.w};
      cacc = __builtin_amdgcn_wmma_f32_16x16x4_f32(false, a0, false, bones,
                                                   (short)0, cacc, false, false);
      cacc = __builtin_amdgcn_wmma_f32_16x16x4_f32(false, a1, false, bones,
                                                   (short)0, cacc, false, false);
    }
  }
  // D columns are identical; per-lane sum of 8 accumulator VGPRs covers
  // M=0..7 (lanes 0-15) / M=8..15 (lanes 16-31) of one column -> one
  // shfl_xor(16) completes the column (= grand total).
  float s = cacc[0] + cacc[1] + cacc[2] + cacc[3] +
            cacc[4] + cacc[5] + cacc[6] + cacc[7];
  s += __shfl_xor(s, 16, 32);
  if (lane == 0) atomicAdd(acc, s);
}

__global__ void k_final(const float* __restrict__ acc, float* __restrict__ out) {
  if (threadIdx.x == 0 && blockIdx.x == 0)
    out[0] = acc[0] * (1.0f / (float)NTOT);
}

// ---------------------------------------------------------------------------
extern "C" void kernel_launch(void* const* d_in, const int* in_sizes, int n_in,
                              void* d_out, int out_size, void* d_ws, size_t ws_size,
                              hipStream_t stream) {
  (void)in_sizes; (void)n_in; (void)out_size; (void)ws_size;
  const float* x = (const float*)d_in[0];
  const float* y = (const float*)d_in[1];

  // workspace layout: res (59MB) | hist1 | hist2 | hist3 | state | acc
  float*    res   = (float*)d_ws;
  unsigned* hist1 = (unsigned*)(res + (size_t)BB * HW);
  unsigned* hist2 = hist1 + BB * 2048;
  unsigned* hist3 = hist2 + BB * 2048;
  unsigned* state = hist3 + BB * 2048;            // 2 uints / batch
  float*    acc   = (float*)(state + 2 * BB + 16);

  size_t zero_bytes = (size_t)((char*)(acc + 1) - (char*)hist1);
  hipMemsetAsync(hist1, 0, zero_bytes, stream);   // graph-capture safe

  dim3 blk(256);
  dim3 grdA(HW / 1024, BB);                       // 900 x 16, 4 px/thread

  k_res_hist<<<grdA, blk, 0, stream>>>(x, y, res, hist1);
  k_scan<<<dim3(BB), dim3(32), 0, stream>>>(hist1, state, 2048, 21, 0);
  k_hist_refine<<<grdA, blk, 0, stream>>>(res, state, hist2, 1);
  k_scan<<<dim3(BB), dim3(32), 0, stream>>>(hist2, state, 2048, 10, 1);
  k_hist_refine<<<grdA, blk, 0, stream>>>(res, state, hist3, 2);
  k_scan<<<dim3(BB), dim3(32), 0, stream>>>(hist3, state, 1024, 0, 2);
  k_masked_sum<<<dim3(HW / (TILE * TILES_PER_BLK), BB), blk, 0, stream>>>(res, state, acc);
  k_final<<<dim3(1), dim3(32), 0, stream>>>(acc, (float*)d_out);
}